// GCNLayer_89764816486619
// MI455X (gfx1250) — compile-verified
//
#include <hip/hip_runtime.h>

typedef __attribute__((ext_vector_type(2))) float v2f;
typedef __attribute__((ext_vector_type(8))) float v8f;

#define D 256              // D_IN == D_OUT
#define BLOCK 256
#define WAVES_PER_BLOCK 8
#define ROWS_PER_WAVE 16
#define COLS_PER_WAVE 64   // 4 WMMA col tiles of 16
#define COL_GROUPS (D / COLS_PER_WAVE)

// ---------------- degree / normalization ----------------

__global__ void deg_init_kernel(float* deg, int N) {
    int i = blockIdx.x * blockDim.x + threadIdx.x;
    if (i < N) deg[i] = 1.0f;  // self loop contributes 1 to each degree
}

__global__ void deg_edge_kernel(const int* __restrict__ src, float* deg, int E) {
    int e = blockIdx.x * blockDim.x + threadIdx.x;
    if (e < E) atomicAdd(&deg[src[e]], 1.0f);
}

__global__ void rsqrt_kernel(float* deg, int N) {
    int i = blockIdx.x * blockDim.x + threadIdx.x;
    if (i < N) deg[i] = rsqrtf(deg[i]);
}

// ---------------- dense projection h = x @ W^T via f32 WMMA ----------------
// grid.x covers row tiles (8 waves/block, 16 rows/wave), grid.y = 4 col groups.
// Each block stages its 64-row slab of W (64 KB) in LDS.

__global__ __launch_bounds__(BLOCK) void gemm_wmma_kernel(
    const float* __restrict__ x, const float* __restrict__ W,
    float* __restrict__ h, int N) {
    __shared__ float lw[COLS_PER_WAVE * D];   // 64 KB

    const int tid = threadIdx.x;
    const int colGroup = blockIdx.y;

    // cooperative LDS fill: W rows [colGroup*64, colGroup*64+64), row-major
    {
        const float4* Wv = (const float4*)(W + (size_t)colGroup * COLS_PER_WAVE * D);
        float4* lv = (float4*)lw;
        for (int i = tid; i < COLS_PER_WAVE * D / 4; i += BLOCK) lv[i] = Wv[i];
    }
    __syncthreads();

    const int wave = tid >> 5;
    const int lane = tid & 31;
    const int half = lane >> 4;          // 0: lanes 0-15 (K,K+1), 1: lanes 16-31 (K+2,K+3)
    const int l16  = lane & 15;

    const int rowBase = (blockIdx.x * WAVES_PER_BLOCK + wave) * ROWS_PER_WAVE;
    int m = rowBase + l16;
    if (m > N - 1) m = N - 1;            // clamp keeps EXEC all-ones through WMMA
    const float* xrow = x + (size_t)m * D + half * 2;
    const float* lb   = lw + l16 * D + half * 2;

    v8f acc0 = {}, acc1 = {}, acc2 = {}, acc3 = {};

    for (int k0 = 0; k0 < D; k0 += 4) {
        // A fragment: 16x4 f32, lane l16 = row, VGPR0/1 = K(+half*2), K+1(+half*2)
        v2f a;  a.x  = xrow[k0];            a.y  = xrow[k0 + 1];
        // B fragments: 4x16 f32 per tile, lane l16 = col, same K split
        v2f b0; b0.x = lb[k0];              b0.y = lb[k0 + 1];
        v2f b1; b1.x = lb[16 * D + k0];     b1.y = lb[16 * D + k0 + 1];
        v2f b2; b2.x = lb[32 * D + k0];     b2.y = lb[32 * D + k0 + 1];
        v2f b3; b3.x = lb[48 * D + k0];     b3.y = lb[48 * D + k0 + 1];

        acc0 = __builtin_amdgcn_wmma_f32_16x16x4_f32(false, a, false, b0, (short)0, acc0, false, false);
        acc1 = __builtin_amdgcn_wmma_f32_16x16x4_f32(false, a, false, b1, (short)0, acc1, false, false);
        acc2 = __builtin_amdgcn_wmma_f32_16x16x4_f32(false, a, false, b2, (short)0, acc2, false, false);
        acc3 = __builtin_amdgcn_wmma_f32_16x16x4_f32(false, a, false, b3, (short)0, acc3, false, false);
    }

    // D layout: VGPR v -> M = rowBase + v + half*8, N = colBase + tile*16 + l16
    const int colBase = colGroup * COLS_PER_WAVE;
    #pragma unroll
    for (int v = 0; v < 8; ++v) {
        int row = rowBase + v + half * 8;
        if (row < N) {
            float* orow = h + (size_t)row * D + colBase + l16;
            orow[0]  = acc0[v];
            orow[16] = acc1[v];
            orow[32] = acc2[v];
            orow[48] = acc3[v];
        }
    }
}

// ---------------- self-loop init: out = dr[i]^2 * h[i] ----------------

__global__ void out_init_kernel(const float* __restrict__ h, const float* __restrict__ dr,
                                float* __restrict__ out, int N) {
    int t = blockIdx.x * blockDim.x + threadIdx.x;     // N*64 threads, float4 each
    if (t >= N * 64) return;
    int i = t >> 6, c = (t & 63) << 2;
    float v = dr[i]; v *= v;
    float4 hv = *(const float4*)(h + (size_t)i * D + c);
    float4 o; o.x = v * hv.x; o.y = v * hv.y; o.z = v * hv.z; o.w = v * hv.w;
    *(float4*)(out + (size_t)i * D + c) = o;
}

// ---------------- edge scatter: out[src] += dr[src]*dr[dst] * h[dst] ----------------
// 64 threads per edge, float4 per thread: fully coalesced 1KB row gathers,
// f32 atomic adds resolve in the 192MB L2 (out fits).

__global__ void scatter_kernel(const int* __restrict__ src, const int* __restrict__ dst,
                               const float* __restrict__ dr, const float* __restrict__ h,
                               float* out, int E) {
    unsigned t = blockIdx.x * blockDim.x + threadIdx.x;  // E*64 threads
    if (t >= (unsigned)E * 64u) return;
    int e = t >> 6, c = (t & 63) << 2;
    int s = src[e], d = dst[e];
    float val = dr[s] * dr[d];
    float4 hv = *(const float4*)(h + (size_t)d * D + c);
    float* o = out + (size_t)s * D + c;
    atomicAdd(o + 0, val * hv.x);
    atomicAdd(o + 1, val * hv.y);
    atomicAdd(o + 2, val * hv.z);
    atomicAdd(o + 3, val * hv.w);
}

// ---------------- launcher ----------------

extern "C" void kernel_launch(void* const* d_in, const int* in_sizes, int n_in,
                              void* d_out, int out_size, void* d_ws, size_t ws_size,
                              hipStream_t stream) {
    const float* x  = (const float*)d_in[0];
    const float* W  = (const float*)d_in[1];
    const int*   ei = (const int*)d_in[2];

    const int N = in_sizes[0] / D;
    const int E = in_sizes[2] / 2;
    const int* src = ei;          // edge_index[0]
    const int* dst = ei + E;      // edge_index[1]

    float* out = (float*)d_out;
    float* h   = (float*)d_ws;                 // N*D floats
    float* deg = h + (size_t)N * D;            // N floats; becomes dr after rsqrt

    deg_init_kernel<<<(N + 255) / 256, 256, 0, stream>>>(deg, N);
    deg_edge_kernel<<<(E + 255) / 256, 256, 0, stream>>>(src, deg, E);
    rsqrt_kernel<<<(N + 255) / 256, 256, 0, stream>>>(deg, N);

    int rowTiles = (N + ROWS_PER_WAVE - 1) / ROWS_PER_WAVE;
    dim3 grid((rowTiles + WAVES_PER_BLOCK - 1) / WAVES_PER_BLOCK, COL_GROUPS);
    gemm_wmma_kernel<<<grid, BLOCK, 0, stream>>>(x, W, h, N);

    long long t1 = (long long)N * 64;
    out_init_kernel<<<(unsigned)((t1 + 255) / 256), 256, 0, stream>>>(h, deg, out, N);

    long long t2 = (long long)E * 64;
    scatter_kernel<<<(unsigned)((t2 + 255) / 256), 256, 0, stream>>>(src, dst, deg, h, out, E);
}